// MessagePassing_43499428773982
// MI455X (gfx1250) — compile-verified
//
#include <hip/hip_runtime.h>

typedef __attribute__((ext_vector_type(2))) float v2f;
typedef __attribute__((ext_vector_type(8))) float v8f;

#define N_HID 18

__device__ __forceinline__ float act_tanh(float x) {
#if __has_builtin(__builtin_amdgcn_tanhf)
  return __builtin_amdgcn_tanhf(x);
#elif __has_builtin(__builtin_amdgcn_tanh_f32)
  return __builtin_amdgcn_tanh_f32(x);
#else
  return tanhf(x);
#endif
}

__device__ __forceinline__ v2f ld2s(const float* p) { return *(const v2f*)p; }

#define WMMA4(a, b, c) \
  __builtin_amdgcn_wmma_f32_16x16x4_f32(false, (a), false, (b), (short)0, (c), false, false)

// ---- LDS layout (float offsets, per block) ----
static constexpr int FRAG1 = 0;            // layer1 B-frags: 2 ksteps * 2 ntiles * 64
static constexpr int FRAG2 = 256;          // layer2: 9 * 2 * 64
static constexpr int FRAG3 = 1408;         // layer3: 14 * 2 * 64
static constexpr int FRAGW = 3200;         // Wr: 90 floats
static constexpr int WAVE0 = 3296;         // per-wave regions start (8B aligned)
static constexpr int HBUF_STRIDE = 108;    // 6 node slots * 18 (slot5 = dup of node0)
static constexpr int MBUF_STRIDE = 90;     // 5 node slots * 18
static constexpr int WAVE_FLOATS = 16 * HBUF_STRIDE + 16 * MBUF_STRIDE;  // 3168
static constexpr int LDS_FLOATS = WAVE0 + 4 * WAVE_FLOATS;               // 15968 -> 63872 B

// Store one 16x16 f32 D-tile (cols tcolBase..tcolBase+15) into an LDS row buffer,
// applying tanh; only columns < 18 are kept. Optionally duplicate into dupOff slot.
__device__ __forceinline__ void store_tile(float* buf, int rowStride, int slotOff,
                                           int tcolBase, int lo, int hi, v8f c,
                                           bool dup, int dupOff) {
  int col = tcolBase + lo;
  if (col < N_HID) {
#pragma unroll
    for (int i = 0; i < 8; ++i) {
      float v = act_tanh(c[i]);
      int m = i + 8 * hi;  // D layout: VGPR i = row i (lanes 0-15) / row i+8 (lanes 16-31)
      buf[m * rowStride + slotOff + col] = v;
      if (dup) buf[m * rowStride + dupOff + col] = v;
    }
  }
}

__global__ void __launch_bounds__(128) ring_mpnn_kernel(
    const float* __restrict__ x, const float* __restrict__ Wf,
    const float* __restrict__ bf, const float* __restrict__ Wm,
    const float* __restrict__ bm, const float* __restrict__ Wu,
    const float* __restrict__ bu, const float* __restrict__ Wr,
    const float* __restrict__ br, float* __restrict__ out, int tiles) {
  extern __shared__ float smem[];
  const int tid = threadIdx.x;

  // ---- cooperative pack: weight B-fragments (WMMA lane layout, zero padded) + Wr ----
  for (int idx = tid; idx < FRAGW + 90; idx += 128) {
    if (idx >= FRAGW) {
      smem[idx] = Wr[idx - FRAGW];
      continue;
    }
    const float* W;
    int K, base;
    if (idx < FRAG2)      { W = Wf; K = 6;  base = FRAG1; }
    else if (idx < FRAG3) { W = Wm; K = 36; base = FRAG2; }
    else                  { W = Wu; K = 54; base = FRAG3; }
    int rem = idx - base;
    int j  = rem & 1;          // which of the lane's 2 floats (VGPR0/VGPR1)
    int ln = (rem >> 1) & 31;  // lane
    int st = rem >> 6;         // (kstep, ntile) group
    int t = st & 1, s = st >> 1;
    // B 4x16 f32 layout: element (k,n) -> VGPR k&1, lane n + 16*(k>>1)
    int ncol = t * 16 + (ln & 15);
    int kk = s * 4 + 2 * (ln >> 4) + j;
    smem[idx] = (ncol < N_HID && kk < K) ? W[kk * N_HID + ncol] : 0.0f;
  }
  __syncthreads();

  const int w = tid >> 5;
  const int lane = tid & 31;
  const int lo = lane & 15;
  const int hi = lane >> 4;
  const int tile = blockIdx.x * 4 + w;
  if (tile >= tiles) return;  // wave-uniform

  const int row0 = tile * 16;
  float* hbuf = smem + WAVE0 + w * WAVE_FLOATS;
  float* mbuf = hbuf + 16 * HBUF_STRIDE;

  // per-lane biases for column tiles 0 (cols 0-15) and 1 (cols 16-17, pad 0)
  const float bf0 = bf[lo], bf1 = (lo < 2) ? bf[16 + lo] : 0.0f;
  const float bm0 = bm[lo], bm1 = (lo < 2) ? bm[16 + lo] : 0.0f;
  const float bu0 = bu[lo], bu1 = (lo < 2) ? bu[16 + lo] : 0.0f;
  const float br0 = br[0];

  // A-fragment rows: lane lo -> batch row0+lo; K half selected by hi
  const float* xrow = x + (size_t)(row0 + lo) * 30;

  // ---------------- Layer 1: h[n] = tanh(x[:,n,:] @ Wf + bf) ----------------
  for (int n = 0; n < 5; ++n) {
    v8f c0 = {bf0, bf0, bf0, bf0, bf0, bf0, bf0, bf0};
    v8f c1 = {bf1, bf1, bf1, bf1, bf1, bf1, bf1, bf1};
#pragma unroll
    for (int s = 0; s < 2; ++s) {
      int kk = s * 4 + 2 * hi;
      v2f a = {0.0f, 0.0f};
      if (kk < 6) a = ld2s(xrow + n * 6 + kk);  // F=6, pad K 6->8
      v2f b0 = ld2s(&smem[FRAG1 + ((s * 2 + 0) * 32 + lane) * 2]);
      v2f b1 = ld2s(&smem[FRAG1 + ((s * 2 + 1) * 32 + lane) * 2]);
      c0 = WMMA4(a, b0, c0);
      c1 = WMMA4(a, b1, c1);
    }
    store_tile(hbuf, HBUF_STRIDE, n * 18, 0,  lo, hi, c0, n == 0, 5 * 18);
    store_tile(hbuf, HBUF_STRIDE, n * 18, 16, lo, hi, c1, n == 0, 5 * 18);
  }

  // ------------- Layer 2: M[n] = tanh([h[n]|h[n+1]] @ Wm + bm), K=36 -------------
  // hbuf slot n..n+1 is the contiguous 36-float concat (slot5 duplicates node0)
  for (int n = 0; n < 5; ++n) {
    v8f c0 = {bm0, bm0, bm0, bm0, bm0, bm0, bm0, bm0};
    v8f c1 = {bm1, bm1, bm1, bm1, bm1, bm1, bm1, bm1};
#pragma unroll
    for (int s = 0; s < 9; ++s) {
      int kk = s * 4 + 2 * hi;
      v2f a  = ld2s(&hbuf[lo * HBUF_STRIDE + n * 18 + kk]);
      v2f b0 = ld2s(&smem[FRAG2 + ((s * 2 + 0) * 32 + lane) * 2]);
      v2f b1 = ld2s(&smem[FRAG2 + ((s * 2 + 1) * 32 + lane) * 2]);
      c0 = WMMA4(a, b0, c0);
      c1 = WMMA4(a, b1, c1);
    }
    store_tile(mbuf, MBUF_STRIDE, n * 18, 0,  lo, hi, c0, false, 0);
    store_tile(mbuf, MBUF_STRIDE, n * 18, 16, lo, hi, c1, false, 0);
  }

  // ---- Layer 3: U[n] = tanh([M[n-1]|h[n]|h[n]] @ Wu + bu), K=54 (pad->56) ----
  // M[n-1] is only read by node n, so U[n] overwrites that slot afterwards.
  for (int n = 0; n < 5; ++n) {
    int rslot = (n + 4) % 5;
    v8f c0 = {bu0, bu0, bu0, bu0, bu0, bu0, bu0, bu0};
    v8f c1 = {bu1, bu1, bu1, bu1, bu1, bu1, bu1, bu1};
#pragma unroll
    for (int s = 0; s < 14; ++s) {
      int kk = s * 4 + 2 * hi;
      // concat boundaries (18,36) are even and lane pairs start even -> no straddle
      int kh = (kk >= 36) ? (kk - 36) : ((kk >= 18) ? (kk - 18) : 0);
      const float* pa = (kk < 18) ? &mbuf[lo * MBUF_STRIDE + rslot * 18 + kk]
                                  : &hbuf[lo * HBUF_STRIDE + n * 18 + kh];
      v2f a  = ld2s(pa);  // kk>=54 reads in-bounds garbage, B-frag is 0 there
      v2f b0 = ld2s(&smem[FRAG3 + ((s * 2 + 0) * 32 + lane) * 2]);
      v2f b1 = ld2s(&smem[FRAG3 + ((s * 2 + 1) * 32 + lane) * 2]);
      c0 = WMMA4(a, b0, c0);
      c1 = WMMA4(a, b1, c1);
    }
    store_tile(mbuf, MBUF_STRIDE, rslot * 18, 0,  lo, hi, c0, false, 0);
    store_tile(mbuf, MBUF_STRIDE, rslot * 18, 16, lo, hi, c1, false, 0);
  }

  // ---------------- Readout: out[b] = U.flatten() @ Wr + br ----------------
  // 2 lanes per row: lane lo handles elements [hi*45, hi*45+45)
  float sum = 0.0f;
  const int e0 = hi * 45;
  for (int j = 0; j < 45; ++j) {
    int e = e0 + j;
    int nn = e / 18;
    int cc = e - nn * 18;
    int slot = (nn + 4) % 5;  // U[nn] lives in mbuf slot (nn-1) mod 5
    sum += mbuf[lo * MBUF_STRIDE + slot * 18 + cc] * smem[FRAGW + e];
  }
  sum += __shfl_xor(sum, 16, 32);
  if (lane < 16) out[row0 + lo] = sum + br0;
}

extern "C" void kernel_launch(void* const* d_in, const int* in_sizes, int n_in,
                              void* d_out, int out_size, void* d_ws, size_t ws_size,
                              hipStream_t stream) {
  (void)n_in; (void)out_size; (void)d_ws; (void)ws_size;
  const float* x  = (const float*)d_in[0];
  const float* Wf = (const float*)d_in[1];
  const float* bf = (const float*)d_in[2];
  const float* Wm = (const float*)d_in[3];
  const float* bm = (const float*)d_in[4];
  const float* Wu = (const float*)d_in[5];
  const float* bu = (const float*)d_in[6];
  const float* Wr = (const float*)d_in[7];
  const float* br = (const float*)d_in[8];
  float* out = (float*)d_out;

  const int B = in_sizes[0] / 30;        // [B,5,6]
  const int tiles = (B + 15) / 16;       // 16 batches per wave
  const int blocks = (tiles + 3) / 4;    // 4 waves (128 threads) per block
  const size_t shmem = (size_t)LDS_FLOATS * sizeof(float);  // 63872 B

  hipLaunchKernelGGL(ring_mpnn_kernel, dim3(blocks), dim3(128), shmem, stream,
                     x, Wf, bf, Wm, bm, Wu, bu, Wr, br, out, tiles);
}